// MeanPositiveSimilarity_35785667510424
// MI455X (gfx1250) — compile-verified
//
#include <hip/hip_runtime.h>

// MI455X / gfx1250, wave32. Memory-bound streaming reduction:
//   67 MB @ 23.3 TB/s => ~2.9 us floor; all design choices serve that.
// WMMA f32 16x16x4 (B = ones) is used as a single-instruction wave-wide
// row-sum engine for the three per-row reductions (qp, qq, pp).

typedef __attribute__((ext_vector_type(2))) float v2f;   // A/B of 16x16x4 f32 WMMA
typedef __attribute__((ext_vector_type(8))) float v8f;   // C/D of 16x16 f32 WMMA

#define N_ROWS 8192
#define D_DIM  1024
#define CHUNKS 8                          // D split 8 ways -> 4096 waves in flight
#define CHUNK_COLS (D_DIM / CHUNKS)       // 128 columns per wave
#define TILES  (N_ROWS / 16)              // 512 row-tiles of 16 rows
#define WAVES_TOTAL (TILES * CHUNKS)      // 4096
#define WAVES_PER_BLOCK 8                 // 256 threads = 8 wave32

// Stage A: each wave reduces a 16-row x 128-col chunk.
// Lane l accumulates only columns of row (l&15); its two accumulators are the
// (K = 2*(l>>4), K+1) slots of the 16x4 A-matrix. One WMMA with B=ones gives
// the full per-row sums, replicated across columns of D.
__global__ __launch_bounds__(256) void mps_partial_kernel(
    const float* __restrict__ q, const float* __restrict__ p,
    float* __restrict__ ws) {
  const int lane  = threadIdx.x & 31;
  const int wave  = blockIdx.x * WAVES_PER_BLOCK + (threadIdx.x >> 5);
  const int tile  = wave >> 3;              // / CHUNKS
  const int chunk = wave & (CHUNKS - 1);
  const int m = lane & 15;                  // row within tile (A-matrix M)
  const int h = lane >> 4;                  // 0 -> K=0,1 ; 1 -> K=2,3
  const int row  = tile * 16 + m;
  // lanes m and m+16 read adjacent float4s of the same row (32B contiguous)
  const int base = row * D_DIM + chunk * CHUNK_COLS + h * 4;
  const float4* __restrict__ q4 = reinterpret_cast<const float4*>(q + base);
  const float4* __restrict__ p4 = reinterpret_cast<const float4*>(p + base);

  float aqp0 = 0.f, aqq0 = 0.f, app0 = 0.f;   // -> A K = 2h
  float aqp1 = 0.f, aqq1 = 0.f, app1 = 0.f;   // -> A K = 2h+1
#pragma unroll
  for (int j = 0; j < 8; ++j) {               // cols base + 8*j .. (+3)
    float4 qv = q4[2 * j];
    float4 pv = p4[2 * j];
    aqp0 += qv.x * pv.x + qv.y * pv.y + qv.z * pv.z + qv.w * pv.w;
    aqq0 += qv.x * qv.x + qv.y * qv.y + qv.z * qv.z + qv.w * qv.w;
    app0 += pv.x * pv.x + pv.y * pv.y + pv.z * pv.z + pv.w * pv.w;
  }
#pragma unroll
  for (int j = 8; j < 16; ++j) {
    float4 qv = q4[2 * j];
    float4 pv = p4[2 * j];
    aqp1 += qv.x * pv.x + qv.y * pv.y + qv.z * pv.z + qv.w * pv.w;
    aqq1 += qv.x * qv.x + qv.y * qv.y + qv.z * qv.z + qv.w * qv.w;
    app1 += pv.x * pv.x + pv.y * pv.y + pv.z * pv.z + pv.w * pv.w;
  }

  // Wave-wide row reduction via WMMA: D[m][*] = sum over the 4 K-slots of row m.
  v2f a_qp = {aqp0, aqp1};
  v2f a_qq = {aqq0, aqq1};
  v2f a_pp = {app0, app1};
  v2f b_one = {1.0f, 1.0f};                 // B = all ones (layout-agnostic)
  v8f c0 = {0.f, 0.f, 0.f, 0.f, 0.f, 0.f, 0.f, 0.f};
  // 8 args: (neg_a, A, neg_b, B, c_mod, C, reuse_a, reuse_b)
  v8f d_qp = __builtin_amdgcn_wmma_f32_16x16x4_f32(false, a_qp, false, b_one,
                                                   (short)0, c0, false, false);
  v8f d_qq = __builtin_amdgcn_wmma_f32_16x16x4_f32(false, a_qq, false, b_one,
                                                   (short)0, c0, false, false);
  v8f d_pp = __builtin_amdgcn_wmma_f32_16x16x4_f32(false, a_pp, false, b_one,
                                                   (short)0, c0, false, false);

  // D layout: VGPR g, lanes 0-15 -> M=g ; lanes 16-31 -> M=g+8 (all N identical).
  // Lane 0 emits rows 0-7, lane 16 emits rows 8-15 of the tile.
  if (m == 0) {
    const int rbase = tile * 16 + h * 8;
#pragma unroll
    for (int g = 0; g < 8; ++g) {
      atomicAdd(&ws[3 * (rbase + g) + 0], d_qp[g]);
      atomicAdd(&ws[3 * (rbase + g) + 1], d_qq[g]);
      atomicAdd(&ws[3 * (rbase + g) + 2], d_pp[g]);
    }
  }
}

// Stage B: one 1024-thread block; normalize per row with eps guard, mean over N.
__global__ __launch_bounds__(1024) void mps_finalize_kernel(
    const float* __restrict__ ws, float* __restrict__ out) {
  __shared__ float red[1024];
  const int t = threadIdx.x;
  float s = 0.f;
  for (int row = t; row < N_ROWS; row += 1024) {
    float qp = ws[3 * row + 0];
    float qq = ws[3 * row + 1];
    float pp = ws[3 * row + 2];
    float denom = fmaxf(sqrtf(qq), 1e-12f) * fmaxf(sqrtf(pp), 1e-12f);
    s += qp / denom;
  }
  red[t] = s;
  __syncthreads();
  for (int off = 512; off > 0; off >>= 1) {
    if (t < off) red[t] += red[t + off];
    __syncthreads();
  }
  if (t == 0) out[0] = red[0] * (1.0f / (float)N_ROWS);
}

extern "C" void kernel_launch(void* const* d_in, const int* in_sizes, int n_in,
                              void* d_out, int out_size, void* d_ws, size_t ws_size,
                              hipStream_t stream) {
  const float* q = (const float*)d_in[0];   // query_logits  [8192,1024] f32
  const float* p = (const float*)d_in[1];   // passage_logits[8192,1024] f32
  float* out = (float*)d_out;               // scalar f32
  float* ws  = (float*)d_ws;                // per-row (qp,qq,pp) accumulators

  // Zero the 96 KB of per-row accumulators (graph-capturable async memset).
  hipMemsetAsync(ws, 0, (size_t)N_ROWS * 3 * sizeof(float), stream);

  mps_partial_kernel<<<WAVES_TOTAL / WAVES_PER_BLOCK, WAVES_PER_BLOCK * 32, 0,
                       stream>>>(q, p, ws);
  mps_finalize_kernel<<<1, 1024, 0, stream>>>(ws, out);
}